// AttentionHead_85521388798159
// MI455X (gfx1250) — compile-verified
//
#include <hip/hip_runtime.h>
#include <hip/hip_bf16.h>

typedef __attribute__((ext_vector_type(2))) float v2f;
typedef __attribute__((ext_vector_type(4))) float v4f;
typedef __attribute__((ext_vector_type(8))) float v8f;

#define EMB    768
#define HS     64
#define NTOK   4096
#define BATCH  4
#define NCHUNK 16                      // K^T V reduction chunks over n
#define CHTOK  (NTOK / NCHUNK)         // 256 tokens per chunk

// ---------------------------------------------------------------------------
// Kernel 1: fused QKV projection. One wave -> one 16-row tile, ALL of Q,K,V.
//   x is read exactly once from HBM.
// WMMA f32 16x16x4. K is processed in groups of 8 with a permuted K-order so
// the A operand is a single b128 load per lane:
//   lane half h loads x[row][k8+4h .. k8+4h+3]
//   wmma1 K-group = {k8,k8+1,k8+4,k8+5}  -> A=(v.x,v.y), B row = k8+4h+r
//   wmma2 K-group = {k8+2,k8+3,k8+6,k8+7}-> A=(v.z,v.w), B row = k8+4h+2+r
// C layout: VGPR r holds C[r+8h][lane&15].
// ---------------------------------------------------------------------------
__global__ __launch_bounds__(256) void qkv_proj_kernel(
    const float* __restrict__ x,
    const float* __restrict__ Wq, const float* __restrict__ bq,
    const float* __restrict__ Wk, const float* __restrict__ bk,
    const float* __restrict__ Wv, const float* __restrict__ bv,
    float* __restrict__ Q, float* __restrict__ K, float* __restrict__ V)
{
    const int wid  = (int)((blockIdx.x * blockDim.x + threadIdx.x) >> 5);
    const int lane = (int)(threadIdx.x & 31);
    const int l = lane & 15;
    const int h = lane >> 4;

    const long rowbase = (long)wid * 16;                  // wid in [0,1024)
    const float* __restrict__ xrow = x + (rowbase + l) * EMB;

    v8f accQ[4] = {}, accK[4] = {}, accV[4] = {};

    for (int k8 = 0; k8 < EMB; k8 += 8) {
        const v4f a = *(const v4f*)(xrow + k8 + 4 * h);
        const int r0 = (k8 + 4 * h) * HS;                 // B row k8+4h
        const int r1 = r0 + HS;                           // k8+4h+1
        const int r2 = r1 + HS;                           // k8+4h+2
        const int r3 = r2 + HS;                           // k8+4h+3
        const v2f a01 = {a.x, a.y};
        const v2f a23 = {a.z, a.w};
#pragma unroll
        for (int t = 0; t < 4; ++t) {
            const int col = t * 16 + l;
            v2f b;
            // ---- Wq ----
            b.x = Wq[r0 + col]; b.y = Wq[r1 + col];
            accQ[t] = __builtin_amdgcn_wmma_f32_16x16x4_f32(
                false, a01, false, b, (short)0, accQ[t], false, false);
            b.x = Wq[r2 + col]; b.y = Wq[r3 + col];
            accQ[t] = __builtin_amdgcn_wmma_f32_16x16x4_f32(
                false, a23, false, b, (short)0, accQ[t], false, false);
            // ---- Wk ----
            b.x = Wk[r0 + col]; b.y = Wk[r1 + col];
            accK[t] = __builtin_amdgcn_wmma_f32_16x16x4_f32(
                false, a01, false, b, (short)0, accK[t], false, false);
            b.x = Wk[r2 + col]; b.y = Wk[r3 + col];
            accK[t] = __builtin_amdgcn_wmma_f32_16x16x4_f32(
                false, a23, false, b, (short)0, accK[t], false, false);
            // ---- Wv ----
            b.x = Wv[r0 + col]; b.y = Wv[r1 + col];
            accV[t] = __builtin_amdgcn_wmma_f32_16x16x4_f32(
                false, a01, false, b, (short)0, accV[t], false, false);
            b.x = Wv[r2 + col]; b.y = Wv[r3 + col];
            accV[t] = __builtin_amdgcn_wmma_f32_16x16x4_f32(
                false, a23, false, b, (short)0, accV[t], false, false);
        }
    }
#pragma unroll
    for (int t = 0; t < 4; ++t) {
        const int col = t * 16 + l;
        const float cq = bq[col], ck = bk[col], cv = bv[col];
#pragma unroll
        for (int r = 0; r < 8; ++r) {
            const long idx = (rowbase + r + 8 * h) * HS + col;
            Q[idx] = accQ[t][r] + cq;
            K[idx] = accK[t][r] + ck;
            V[idx] = accV[t][r] + cv;
        }
    }
}

// ---------------------------------------------------------------------------
// Kernel 2a: partial S = K^T V over a 256-token chunk.
//   P[b][chunk][64][64] ; one wave per (tile, chunk, batch); 64 K-iterations.
// A[m][c] = K[c][m] ; B[c][n] = V[c][n]  (both stride-HS loads, b32)
// ---------------------------------------------------------------------------
__global__ __launch_bounds__(256) void ktv_partial_kernel(
    const float* __restrict__ K, const float* __restrict__ V,
    float* __restrict__ P)
{
    const int wave = (int)(threadIdx.x >> 5);
    const int lane = (int)(threadIdx.x & 31);
    const int tile  = (int)blockIdx.x * 8 + wave;    // 0..15
    const int chunk = (int)blockIdx.y;               // 0..15
    const int b     = (int)blockIdx.z;
    const int mt = tile & 3, nt = tile >> 2;
    const int l = lane & 15, h = lane >> 4;

    const float* __restrict__ Kb = K + (long)b * NTOK * HS + (long)chunk * CHTOK * HS;
    const float* __restrict__ Vb = V + (long)b * NTOK * HS + (long)chunk * CHTOK * HS;

    v8f acc = {};
    for (int kk = 0; kk < CHTOK; kk += 4) {
        const int c = kk + 2 * h;
        v2f a, bb;
        a.x  = Kb[(long)c       * HS + mt * 16 + l];
        a.y  = Kb[(long)(c + 1) * HS + mt * 16 + l];
        bb.x = Vb[(long)c       * HS + nt * 16 + l];
        bb.y = Vb[(long)(c + 1) * HS + nt * 16 + l];
        acc = __builtin_amdgcn_wmma_f32_16x16x4_f32(
            false, a, false, bb, (short)0, acc, false, false);
    }
    float* __restrict__ Pb = P + ((long)b * NCHUNK + chunk) * HS * HS;
#pragma unroll
    for (int r = 0; r < 8; ++r) {
        Pb[(mt * 16 + r + 8 * h) * HS + nt * 16 + l] = acc[r];
    }
}

// ---------------------------------------------------------------------------
// Kernel 2b: deterministic fixed-order reduction of partials:
//   S[b][i][j] = sum_c P[b][c][i][j]
// ---------------------------------------------------------------------------
__global__ __launch_bounds__(256) void ktv_reduce_kernel(
    const float* __restrict__ P, float* __restrict__ S)
{
    const int e = (int)(blockIdx.x * blockDim.x + threadIdx.x); // 0..B*64*64-1
    const int b = e / (HS * HS);
    const int ij = e % (HS * HS);
    const float* __restrict__ Pb = P + (long)b * NCHUNK * HS * HS;
    float s = 0.0f;
#pragma unroll
    for (int c = 0; c < NCHUNK; ++c) s += Pb[(long)c * HS * HS + ij];
    S[e] = s;
}

// ---------------------------------------------------------------------------
// Kernel 3: out[b] = scale * Q[b] @ S[b]   ([4096x64] . [64x64])
// One wave per 16x64 row tile; b128 A loads with the permuted-K trick.
// ---------------------------------------------------------------------------
__global__ __launch_bounds__(256) void qs_kernel(
    const float* __restrict__ Q, const float* __restrict__ S,
    float* __restrict__ out, const float scale)
{
    const int wave = (int)(threadIdx.x >> 5);
    const int lane = (int)(threadIdx.x & 31);
    const int rowtile = (int)blockIdx.x * 8 + wave;   // 0..255
    const int b = (int)blockIdx.y;
    const int l = lane & 15, h = lane >> 4;

    const float* __restrict__ Qb = Q + (long)b * NTOK * HS;
    const float* __restrict__ Sb = S + b * HS * HS;
    float* __restrict__ Ob       = out + (long)b * NTOK * HS;

    const long rowbase = (long)rowtile * 16;
    const float* __restrict__ qrow = Qb + (rowbase + l) * HS;

    v8f acc[4] = {};
#pragma unroll
    for (int k8 = 0; k8 < HS; k8 += 8) {
        const v4f a = *(const v4f*)(qrow + k8 + 4 * h);
        const int r0 = (k8 + 4 * h) * HS;
        const int r1 = r0 + HS;
        const int r2 = r1 + HS;
        const int r3 = r2 + HS;
        const v2f a01 = {a.x, a.y};
        const v2f a23 = {a.z, a.w};
#pragma unroll
        for (int t = 0; t < 4; ++t) {
            const int col = t * 16 + l;
            v2f bb;
            bb.x = Sb[r0 + col]; bb.y = Sb[r1 + col];
            acc[t] = __builtin_amdgcn_wmma_f32_16x16x4_f32(
                false, a01, false, bb, (short)0, acc[t], false, false);
            bb.x = Sb[r2 + col]; bb.y = Sb[r3 + col];
            acc[t] = __builtin_amdgcn_wmma_f32_16x16x4_f32(
                false, a23, false, bb, (short)0, acc[t], false, false);
        }
    }
#pragma unroll
    for (int t = 0; t < 4; ++t) {
#pragma unroll
        for (int r = 0; r < 8; ++r) {
            Ob[(rowbase + r + 8 * h) * HS + t * 16 + l] = acc[t][r] * scale;
        }
    }
}

extern "C" void kernel_launch(void* const* d_in, const int* in_sizes, int n_in,
                              void* d_out, int out_size, void* d_ws, size_t ws_size,
                              hipStream_t stream)
{
    const float* x  = (const float*)d_in[0];
    const float* Wq = (const float*)d_in[1];
    const float* bq = (const float*)d_in[2];
    const float* Wk = (const float*)d_in[3];
    const float* bk = (const float*)d_in[4];
    const float* Wv = (const float*)d_in[5];
    const float* bv = (const float*)d_in[6];
    float* out = (float*)d_out;

    // Workspace layout: Q | K | V | P (partials) | S
    const size_t qkv_elems = (size_t)BATCH * NTOK * HS;      // 1M floats each
    float* Q = (float*)d_ws;
    float* K = Q + qkv_elems;
    float* V = K + qkv_elems;
    float* P = V + qkv_elems;                                 // B*16*64*64
    float* S = P + (size_t)BATCH * NCHUNK * HS * HS;          // B*64*64

    const float scale = 1.0f / sqrtf((float)EMB);

    // 1) Fused QKV projection: 1024 waves, 8 waves/block -> 128 blocks
    qkv_proj_kernel<<<dim3(128), dim3(256), 0, stream>>>(
        x, Wq, bq, Wk, bk, Wv, bv, Q, K, V);

    // 2a) Partial K^T V: (2 blocks x 8 tiles) x 16 chunks x B
    ktv_partial_kernel<<<dim3(2, NCHUNK, BATCH), dim3(256), 0, stream>>>(K, V, P);

    // 2b) Deterministic reduction: B*64*64 = 16384 elements
    ktv_reduce_kernel<<<dim3((BATCH * HS * HS) / 256), dim3(256), 0, stream>>>(P, S);

    // 3) out = scale * Q @ S: 256 row tiles/batch, 8 waves/block -> (32, B)
    qs_kernel<<<dim3(32, BATCH), dim3(256), 0, stream>>>(Q, S, out, scale);
}